// WiringEntropyRegulariser_40450001993761
// MI455X (gfx1250) — compile-verified
//
#include <hip/hip_runtime.h>
#include <stdint.h>

#define NUM_BINS 30
#define LAMBD    0.01f
#define EPSF     1e-8f

#define BLOCK       256          // 8 waves (wave32)
#define NWAVES      (BLOCK / 32)
#define TILE        1024         // floats per tile (per array) = 4 KB
#define HIST_BLOCKS 2048
#define MM_BLOCKS   1024

// d_ws layout (32-bit words):
//   [0] : min key (uint, monotone float encoding)
//   [1] : max key (uint)
//   [2..33] : 32 float bin accumulators (bins 0..29 used)

__device__ __forceinline__ unsigned f2key(float f) {
  unsigned u = __float_as_uint(f);
  return (u & 0x80000000u) ? ~u : (u | 0x80000000u);
}
__device__ __forceinline__ float key2f(unsigned k) {
  unsigned u = (k & 0x80000000u) ? (k & 0x7FFFFFFFu) : ~k;
  return __uint_as_float(u);
}

// ---------------------------------------------------------------- init ----
__global__ void k_init(unsigned* __restrict__ ws) {
  int t = threadIdx.x;
  if (t == 0) ws[0] = 0xFFFFFFFFu;          // +inf key
  if (t == 1) ws[1] = 0x00000000u;          // -inf key
  if (t >= 2 && t < 34) ((float*)ws)[t] = 0.0f;
}

// -------------------------------------------------------------- minmax ----
__global__ void k_minmax(const float* __restrict__ d, unsigned* __restrict__ ws,
                         long long n) {
  const int tid = threadIdx.x;
  const long long stride = (long long)gridDim.x * BLOCK;
  long long gid = (long long)blockIdx.x * BLOCK + tid;

  float vmin = __uint_as_float(0x7F800000u);   // +inf
  float vmax = __uint_as_float(0xFF800000u);   // -inf

  const long long n4 = n >> 2;
  const float4* __restrict__ d4 = (const float4*)d;
  const long long guard = (n4 > stride) ? (n4 - stride) : 0;
  for (long long i = gid; i < n4; i += stride) {
    if (i < guard) __builtin_prefetch(&d4[i + stride], 0, 1);
    float4 v = d4[i];
    vmin = fminf(fminf(vmin, v.x), fminf(v.y, fminf(v.z, v.w)));
    vmax = fmaxf(fmaxf(vmax, v.x), fmaxf(v.y, fmaxf(v.z, v.w)));
  }
  for (long long i = (n4 << 2) + gid; i < n; i += stride) {   // tail
    float v = d[i];
    vmin = fminf(vmin, v);
    vmax = fmaxf(vmax, v);
  }

  // wave32 reduction
  #pragma unroll
  for (int off = 16; off > 0; off >>= 1) {
    vmin = fminf(vmin, __shfl_down(vmin, off, 32));
    vmax = fmaxf(vmax, __shfl_down(vmax, off, 32));
  }

  __shared__ float smin[NWAVES];
  __shared__ float smax[NWAVES];
  if ((tid & 31) == 0) { smin[tid >> 5] = vmin; smax[tid >> 5] = vmax; }
  __syncthreads();
  if (tid == 0) {
    #pragma unroll
    for (int wv = 1; wv < NWAVES; ++wv) {
      vmin = fminf(vmin, smin[wv]);
      vmax = fmaxf(vmax, smax[wv]);
    }
    atomicMin(&ws[0], f2key(vmin));
    atomicMax(&ws[1], f2key(vmax));
  }
}

// ------------------------------------------------------------ histogram ----
__global__ void k_hist(const float* __restrict__ w, const float* __restrict__ dm,
                       const unsigned* __restrict__ ws, float* __restrict__ bins,
                       long long n, int numTiles) {
  __shared__ float4 stg_d[2][BLOCK];
  __shared__ float4 stg_w[2][BLOCK];
  __shared__ float  shist[NWAVES][32];

  const int tid = threadIdx.x;
  const int wv  = tid >> 5;
  shist[wv][tid & 31] = 0.0f;   // each wave zeroes exactly its own row

  const float mn    = key2f(ws[0]);
  const float mx    = key2f(ws[1]);
  const float rng   = mx - mn;
  const float step  = rng * (1.0f / (float)NUM_BINS);
  const float scale = (rng > 0.0f) ? ((float)NUM_BINS / rng) : 0.0f;

  // one async b128 per lane per array: stage tile into LDS (ASYNCcnt path)
  auto issue = [&](int tile, int b) {
    long long eoff = (long long)tile * TILE + (long long)tid * 4;
    unsigned  ld = (unsigned)(uintptr_t)(&stg_d[b][tid]);
    unsigned  lw = (unsigned)(uintptr_t)(&stg_w[b][tid]);
    uint64_t  gd = (uint64_t)(uintptr_t)(dm + eoff);
    uint64_t  gw = (uint64_t)(uintptr_t)(w  + eoff);
    asm volatile("global_load_async_to_lds_b128 %0, %1, off"
                 :: "v"(ld), "v"(gd) : "memory");
    asm volatile("global_load_async_to_lds_b128 %0, %1, off"
                 :: "v"(lw), "v"(gw) : "memory");
  };

  // idx = searchsorted(linspace(mn,mx,31), x, 'left') = #edges < x
  //   idx==0  -> dropped (overflow bin)
  //   x==max  -> idx==30 -> bin 29  (clamp handles fp rounding of last edge)
  auto process = [&](float x, float wval) {
    float aw = fabsf(wval);
    float f  = (x - mn) * scale;
    int   a  = (int)f;
    a = a < 0 ? 0 : (a > NUM_BINS ? NUM_BINS : a);
    if (a <= NUM_BINS && fmaf((float)a,       step, mn) <  x) ++a;
    if (a <= NUM_BINS && fmaf((float)a,       step, mn) <  x) ++a;
    if (a > 0         && fmaf((float)(a - 1), step, mn) >= x) --a;
    if (a > NUM_BINS) a = NUM_BINS;        // linspace endpoint is exactly max
    int seg = (a >= 1) ? (a - 1) : NUM_BINS;
    if (seg < NUM_BINS) atomicAdd(&shist[wv][seg], aw);
  };

  int t   = blockIdx.x;
  int buf = 0;
  if (t < numTiles) issue(t, buf);

  for (; t < numTiles; t += gridDim.x) {
    int  tn   = t + gridDim.x;
    bool more = tn < numTiles;
    if (more) {
      issue(tn, buf ^ 1);
      asm volatile("s_wait_asynccnt 2" ::: "memory");  // current tile landed
    } else {
      asm volatile("s_wait_asynccnt 0" ::: "memory");
    }
    float4 dv = stg_d[buf][tid];
    float4 wq = stg_w[buf][tid];
    process(dv.x, wq.x);
    process(dv.y, wq.y);
    process(dv.z, wq.z);
    process(dv.w, wq.w);
    buf ^= 1;
  }

  // tail elements not covered by full tiles
  const long long base = (long long)numTiles * TILE;
  for (long long i = base + (long long)blockIdx.x * BLOCK + tid; i < n;
       i += (long long)gridDim.x * BLOCK) {
    process(dm[i], w[i]);
  }

  __syncthreads();
  if (tid < NUM_BINS) {
    float s = 0.0f;
    #pragma unroll
    for (int r = 0; r < NWAVES; ++r) s += shist[r][tid];
    atomicAdd(&bins[tid], s);
  }
}

// ------------------------------------------------------------- finalize ----
__global__ void k_final(const float* __restrict__ bins, float* __restrict__ out) {
  if (threadIdx.x == 0) {
    float total = 0.0f;
    #pragma unroll
    for (int b = 0; b < NUM_BINS; ++b) total += bins[b];
    total += EPSF;
    float ne = 0.0f;
    #pragma unroll
    for (int b = 0; b < NUM_BINS; ++b) {
      float p = bins[b] / total;
      ne += p * logf(p + EPSF);
    }
    out[0] = LAMBD * ne;
  }
}

// --------------------------------------------------------------- launch ----
extern "C" void kernel_launch(void* const* d_in, const int* in_sizes, int n_in,
                              void* d_out, int out_size, void* d_ws, size_t ws_size,
                              hipStream_t stream) {
  const float* w_hh = (const float*)d_in[0];
  const float* dmat = (const float*)d_in[1];
  long long n = (long long)in_sizes[0];

  unsigned* ws   = (unsigned*)d_ws;
  float*    bins = (float*)(ws + 2);

  k_init<<<1, 64, 0, stream>>>(ws);
  k_minmax<<<MM_BLOCKS, BLOCK, 0, stream>>>(dmat, ws, n);

  int numTiles = (int)(n / TILE);
  k_hist<<<HIST_BLOCKS, BLOCK, 0, stream>>>(w_hh, dmat, ws, bins, n, numTiles);
  k_final<<<1, 32, 0, stream>>>(bins, (float*)d_out);
}